// FourierResistor_71880572666164
// MI455X (gfx1250) — compile-verified
//
#include <hip/hip_runtime.h>

#define BB 8
#define NN 4096
#define DD 128
#define HH 4
#define BN (BB*NN)
#define NSPLIT 32

typedef __attribute__((ext_vector_type(2))) float v2f;
typedef __attribute__((ext_vector_type(8))) float v8f;

// ---------------------------------------------------------------------------
// Generic Y[M x DD] = epi(A[M x 128] @ W[DD x 128]^T + bias)
// EPI: 0 = none, 1 = relu(v + bias), 2 = v + bias
// One wave per 16x64 output strip: 4 independent accumulator chains,
// A fragment amortized over 4 WMMAs per k-step.
// ---------------------------------------------------------------------------
template <int EPI>
__global__ __launch_bounds__(256) void gemm_d_kernel(const float* __restrict__ A,
                                                     const float* __restrict__ W,
                                                     const float* __restrict__ bias,
                                                     float* __restrict__ Y) {
    const int K = DD;
    const int CT = 4;                       // col tiles per wave
    const int WPS = (DD / 16) / CT;         // waves per row strip = 2
    const int wid = blockIdx.x * 8 + (threadIdx.x >> 5);
    const int m0 = (wid / WPS) * 16;
    const int j0 = (wid % WPS) * (CT * 16);
    const int lane = threadIdx.x & 31;
    const int half = lane >> 4;
    const int lr = lane & 15;

    const float* ap = A + (size_t)(m0 + lr) * K + 2 * half;
    const float* bp = W + (size_t)(j0 + lr) * K + 2 * half;

    v8f acc[CT] = {};
#pragma unroll 4
    for (int k = 0; k < K; k += 4) {
        v2f a = *(const v2f*)(ap + k);
#pragma unroll
        for (int c = 0; c < CT; ++c) {
            v2f b = *(const v2f*)(bp + (size_t)c * 16 * K + k);
            acc[c] = __builtin_amdgcn_wmma_f32_16x16x4_f32(false, a, false, b,
                                                           (short)0, acc[c], false, false);
        }
    }

#pragma unroll
    for (int c = 0; c < CT; ++c) {
        const int col = j0 + c * 16 + lr;
        const float bv = EPI ? bias[col] : 0.f;
#pragma unroll
        for (int r = 0; r < 8; ++r) {
            const int row = m0 + r + 8 * half;
            float v = acc[c][r];
            if (EPI) v += bv;
            if (EPI == 1) v = fmaxf(v, 0.f);
            Y[(size_t)row * DD + col] = v;
        }
    }
}

// ---------------------------------------------------------------------------
// K2a: partial Fourier sums over an N-split (coalesced: thread = dd).
//   Spart[split][0][b][dd] = sum_n xw[b,n,dd]*pos[n,(dd&~1)+1]
//   Spart[split][1][b][dd] = sum_n xw[b,n,dd]*(-pos[n,dd&~1])
// ---------------------------------------------------------------------------
__global__ __launch_bounds__(128) void fsum_part_kernel(const float* __restrict__ xw,
                                                        const float* __restrict__ pos,
                                                        float* __restrict__ Spart) {
    const int b = blockIdx.x >> 5;          // / NSPLIT
    const int split = blockIdx.x & (NSPLIT - 1);
    const int dd = threadIdx.x;
    const int base = dd & ~1;
    const int n0 = split * (NN / NSPLIT);
    float s0 = 0.f, s1 = 0.f;
    for (int i = 0; i < NN / NSPLIT; ++i) {
        const int n = n0 + i;
        const float xv = xw[((size_t)b * NN + n) * DD + dd];
        s0 += xv * pos[(size_t)n * DD + base + 1];
        s1 -= xv * pos[(size_t)n * DD + base];
    }
    float* sp = Spart + (size_t)split * (2 * BB * DD);
    sp[b * DD + dd] = s0;
    sp[BB * DD + b * DD + dd] = s1;
}

// ---------------------------------------------------------------------------
// Deterministic finalize: out[i] = sum_{p<NSPLIT} part[p*stride + i]
// ---------------------------------------------------------------------------
__global__ __launch_bounds__(256) void reduce_parts_kernel(const float* __restrict__ part,
                                                           float* __restrict__ out,
                                                           int n, int stride) {
    const int i = blockIdx.x * 256 + threadIdx.x;
    if (i < n) {
        float s = 0.f;
        for (int p = 0; p < NSPLIT; ++p) s += part[(size_t)p * stride + i];
        out[i] = s;
    }
}

// ---------------------------------------------------------------------------
// K3: phi via WMMA (A = xw + pos broadcast over batch), 4 col tiles per wave,
// fused resistor epilogue:
//     f0 = S0 - xw*ee0 ; f1 = S1 + xw*ee1 ; fn = |f| ; fec = f0*ee0 - f1*ee1
//     rfe = relu(min(1, fn - phi)) * fec                     [BN, H, DD]
// ---------------------------------------------------------------------------
__global__ __launch_bounds__(256) void phi_rfe_kernel(const float* __restrict__ xw,
                                                      const float* __restrict__ pos,
                                                      const float* __restrict__ W_phi,
                                                      const float* __restrict__ b_phi,
                                                      const float* __restrict__ S,
                                                      float* __restrict__ rfe) {
    const int K = DD;
    const int CT = 4;
    const int WPS = (HH * DD / 16) / CT;    // 8 waves per row strip
    const int wid = blockIdx.x * 8 + (threadIdx.x >> 5);
    const int m0 = (wid / WPS) * 16;
    const int j0 = (wid % WPS) * (CT * 16);
    const int lane = threadIdx.x & 31;
    const int half = lane >> 4;
    const int lr = lane & 15;

    const int arow = m0 + lr;
    const int an = arow & (NN - 1);
    const float* ap = xw + (size_t)arow * K + 2 * half;
    const float* pp = pos + (size_t)an * K + 2 * half;
    const float* bp = W_phi + (size_t)(j0 + lr) * K + 2 * half;

    v8f acc[CT] = {};
#pragma unroll 4
    for (int k = 0; k < K; k += 4) {
        v2f a = *(const v2f*)(ap + k);
        v2f p = *(const v2f*)(pp + k);
        a = a + p;
#pragma unroll
        for (int c = 0; c < CT; ++c) {
            v2f b = *(const v2f*)(bp + (size_t)c * 16 * K + k);
            acc[c] = __builtin_amdgcn_wmma_f32_16x16x4_f32(false, a, false, b,
                                                           (short)0, acc[c], false, false);
        }
    }

#pragma unroll
    for (int c = 0; c < CT; ++c) {
        const int col = j0 + c * 16 + lr;   // j in [0, 512)
        const int h = col >> 7;
        const int dd = col & (DD - 1);
        const int base = dd & ~1;
        const float bphi = b_phi[col];
#pragma unroll
        for (int r = 0; r < 8; ++r) {
            const int row = m0 + r + 8 * half;   // m in [0, BN)
            const int b = row >> 12;             // / NN
            const int n = row & (NN - 1);
            const float ph = acc[c][r] + bphi;
            const float xv = xw[(size_t)row * DD + dd];
            const float ee0 = pos[(size_t)n * DD + base + 1];
            const float ee1 = pos[(size_t)n * DD + base];
            const float f0 = S[b * DD + dd] - xv * ee0;
            const float f1 = S[BB * DD + b * DD + dd] + xv * ee1;
            const float fn = sqrtf(f0 * f0 + f1 * f1);
            const float fec = f0 * ee0 - f1 * ee1;
            const float res = fmaxf(fminf(1.f, fn - ph), 0.f);
            rfe[((size_t)row * HH + h) * DD + dd] = res * fec;
        }
    }
}

// ---------------------------------------------------------------------------
// K4a: Tpart[split][b][hd] = sum over N-split of rfe[b,n,hd]
// Coalesced: thread = contiguous hd; serial loop over its n's.
// ---------------------------------------------------------------------------
__global__ __launch_bounds__(256) void tsum_part_kernel(const float* __restrict__ rfe,
                                                        float* __restrict__ Tpart) {
    const int b = blockIdx.x >> 6;              // 64 blocks per b (2 chunks * 32 splits)
    const int chunk = (blockIdx.x >> 5) & 1;
    const int split = blockIdx.x & (NSPLIT - 1);
    const int hd = chunk * 256 + threadIdx.x;   // [0, 512)
    const int n0 = split * (NN / NSPLIT);
    float s = 0.f;
    for (int i = 0; i < NN / NSPLIT; ++i)
        s += rfe[((size_t)b * NN + n0 + i) * (HH * DD) + hd];
    Tpart[(size_t)split * (BB * HH * DD) + b * (HH * DD) + hd] = s;
}

// ---------------------------------------------------------------------------
// K5: per (b,n) row of 128 lanes: inv -> comb (4x4 & 1x4) -> m = xw + comb
//     -> LayerNorm over d -> write mn in place over xw.
// ---------------------------------------------------------------------------
__global__ __launch_bounds__(128) void comb_ln_kernel(float* __restrict__ xw_mn,
                                                      const float* __restrict__ rfe,
                                                      const float* __restrict__ T,
                                                      const float* __restrict__ W_c1,
                                                      const float* __restrict__ b_c1,
                                                      const float* __restrict__ W_c2,
                                                      const float* __restrict__ b_c2,
                                                      const float* __restrict__ ln_g,
                                                      const float* __restrict__ ln_b) {
    __shared__ float sm[128];
    const int row = blockIdx.x;            // b*NN + n
    const int b = row >> 12;
    const int dd = threadIdx.x;

    float t[HH];
#pragma unroll
    for (int h = 0; h < HH; ++h) {
        const float r = rfe[((size_t)row * HH + h) * DD + dd];
        t[h] = (T[(b * HH + h) * DD + dd] - r) * (1.f / NN);
    }
    float comb = b_c2[0];
#pragma unroll
    for (int i = 0; i < HH; ++i) {
        float u = b_c1[i];
#pragma unroll
        for (int j = 0; j < HH; ++j) u += W_c1[i * HH + j] * t[j];
        comb += W_c2[i] * fmaxf(u, 0.f);
    }

    const float m = xw_mn[(size_t)row * DD + dd] + comb;

    sm[dd] = m;
    __syncthreads();
    for (int st = 64; st > 0; st >>= 1) {
        if (dd < st) sm[dd] += sm[dd + st];
        __syncthreads();
    }
    const float mu = sm[0] * (1.f / DD);
    __syncthreads();
    const float dm = m - mu;
    sm[dd] = dm * dm;
    __syncthreads();
    for (int st = 64; st > 0; st >>= 1) {
        if (dd < st) sm[dd] += sm[dd + st];
        __syncthreads();
    }
    const float var = sm[0] * (1.f / DD);
    const float mn = dm * rsqrtf(var + 1e-6f) * ln_g[dd] + ln_b[dd];
    xw_mn[(size_t)row * DD + dd] = mn;
}

// ---------------------------------------------------------------------------
extern "C" void kernel_launch(void* const* d_in, const int* in_sizes, int n_in,
                              void* d_out, int out_size, void* d_ws, size_t ws_size,
                              hipStream_t stream) {
    const float* x     = (const float*)d_in[0];
    const float* pos   = (const float*)d_in[1];
    const float* W_lt  = (const float*)d_in[2];
    const float* W_phi = (const float*)d_in[3];
    const float* b_phi = (const float*)d_in[4];
    const float* W_c1  = (const float*)d_in[5];
    const float* b_c1  = (const float*)d_in[6];
    const float* W_c2  = (const float*)d_in[7];
    const float* b_c2  = (const float*)d_in[8];
    const float* ln_g  = (const float*)d_in[9];
    const float* ln_b  = (const float*)d_in[10];
    const float* W_f1  = (const float*)d_in[11];
    const float* b_f1  = (const float*)d_in[12];
    const float* W_f2  = (const float*)d_in[13];
    const float* b_f2  = (const float*)d_in[14];
    float* out = (float*)d_out;

    float* ws    = (float*)d_ws;
    float* xw    = ws;                                 // BN*DD (also mn, in place)
    float* rfe   = xw + (size_t)BN * DD;               // BN*HH*DD (also hidden later)
    float* S     = rfe + (size_t)BN * HH * DD;         // 2*BB*DD
    float* T     = S + 2 * BB * DD;                    // BB*HH*DD
    float* Spart = T + BB * HH * DD;                   // NSPLIT * 2*BB*DD
    float* Tpart = Spart + (size_t)NSPLIT * 2 * BB * DD; // NSPLIT * BB*HH*DD

    const int gemm_blocks = (BN / 16) * 2 / 8;         // 4096 waves / 8 per block

    // K1: xw = x @ W_lt^T
    gemm_d_kernel<0><<<gemm_blocks, 256, 0, stream>>>(x, W_lt, nullptr, xw);
    // K2: Fourier sums over nodes (split partials + deterministic finalize)
    fsum_part_kernel<<<BB * NSPLIT, 128, 0, stream>>>(xw, pos, Spart);
    reduce_parts_kernel<<<(2 * BB * DD) / 256, 256, 0, stream>>>(Spart, S,
                                                                 2 * BB * DD, 2 * BB * DD);
    // K3: phi GEMM + fused resistor epilogue -> rfe
    phi_rfe_kernel<<<(BN / 16) * 8 / 8, 256, 0, stream>>>(xw, pos, W_phi, b_phi, S, rfe);
    // K4: T = sum_n rfe (split partials + deterministic finalize)
    tsum_part_kernel<<<BB * 2 * NSPLIT, 256, 0, stream>>>(rfe, Tpart);
    reduce_parts_kernel<<<(BB * HH * DD) / 256, 256, 0, stream>>>(Tpart, T,
                                                                  BB * HH * DD, BB * HH * DD);
    // K5: inverse transform + comb MLP + LayerNorm (writes mn over xw)
    comb_ln_kernel<<<BN, 128, 0, stream>>>(xw, rfe, T, W_c1, b_c1, W_c2, b_c2, ln_g, ln_b);
    // K6: hidden = relu(mn @ W_f1^T + b_f1)   (hidden aliases rfe storage)
    gemm_d_kernel<1><<<gemm_blocks, 256, 0, stream>>>(xw, W_f1, b_f1, rfe);
    // K7: out = hidden @ W_f2^T + b_f2
    gemm_d_kernel<2><<<gemm_blocks, 256, 0, stream>>>(rfe, W_f2, b_f2, out);
}